// MyLeNetMatStochBU_76673756168678
// MI455X (gfx1250) — compile-verified
//
#include <hip/hip_runtime.h>

typedef __attribute__((ext_vector_type(2))) float v2f;
typedef __attribute__((ext_vector_type(8))) float v8f;

// K padded up to a multiple of 4 (WMMA f32 step).
__host__ __device__ constexpr int kpFor(int k) { return (k + 3) & ~3; }
// K-chunk staged in LDS; chosen to divide each layer's KP exactly and stay
// under ~40 KB of LDS (KP values here: 76, 1800, 3600, 800).
__host__ __device__ constexpr int kcFor(int kp) {
  return (kp <= 600) ? kp
                     : ((kp % 600 == 0) ? 600 : ((kp % 400 == 0) ? 400 : 512));
}
__host__ __device__ constexpr int smemFor(int k) {
  return 16 * (kcFor(kpFor(k)) + 4) * (int)sizeof(float);
}

// Zero-pad W (Cout x K) into Wp (CoutP x KP) so GEMM A-feeds need no guards.
__global__ __launch_bounds__(256) void pad_weights(
    const float* __restrict__ src, float* __restrict__ dst, int Cout, int K,
    int CoutP, int KP) {
  int idx = blockIdx.x * blockDim.x + threadIdx.x;
  if (idx >= CoutP * KP) return;
  int o = idx / KP;
  int k = idx - o * KP;
  dst[idx] = (o < Cout && k < K) ? src[(size_t)o * K + k] : 0.0f;
}

// One 256-thread block (8 wave32) owns one (spatial position, 16-batch) output
// column. The im2col patch chunk is staged once into LDS and shared by all
// Cout tiles; each wave accumulates ceil(MT/8) 16x16 tiles with
// V_WMMA_F32_16X16X4_F32 (fp32 in / fp32 accum -> reference precision).
//
// A (16x4 f32, ISA 7.12.2): lane<16 -> M=lane, K={k,k+1}; lane>=16 -> M=lane-16,
// K={k+2,k+3}. B (4x16): mirrored with N = lane&15. C/D: vgpr r -> M=r+8*half.
template <int CIN, int HIN, int WIN, int COUT, int KSZ, int STRIDE, int HO,
          int WO, int RELU>
__global__ __launch_bounds__(256) void sconv_wmma(
    const float* __restrict__ in, const float* __restrict__ Wp,
    const float* __restrict__ bias, const int* __restrict__ selh,
    const int* __restrict__ selw, const float* __restrict__ mask,
    float* __restrict__ out) {
  constexpr int K = CIN * KSZ * KSZ;
  constexpr int K2 = KSZ * KSZ;
  constexpr int KP = kpFor(K);
  constexpr int KC = kcFor(KP);
  constexpr int NCHUNK = (KP + KC - 1) / KC;
  constexpr int HC = HIN - KSZ + 1;
  constexpr int WC = WIN - KSZ + 1;
  constexpr int MT = (COUT + 15) / 16;      // Cout tiles of 16
  constexpr int NW = 8;                     // waves per block
  constexpr int TPW = (MT + NW - 1) / NW;   // tiles per wave
  constexpr int LROW = KC + 4;              // LDS pitch (dwords), bank-skewed
  constexpr int HW = HO * WO;

  extern __shared__ float lds[];            // 16 * LROW floats

  const int pos = blockIdx.x;
  const int nt = blockIdx.y;
  const int xo = pos / WO;
  const int yo = pos - xo * WO;

  const int tid = threadIdx.x;
  const int lane = tid & 31;
  const int wave = tid >> 5;
  const int half = lane >> 4;
  const int l16 = lane & 15;
  const int b = nt * 16 + l16;              // batch column this lane stores

  const float mval = mask ? mask[pos] : 1.0f;  // uniform per block
  if (mval == 0.0f) {
    // Reference output is exactly zero at masked positions: skip the GEMM.
    for (int i = tid; i < COUT * 16; i += 256) {
      int o = i >> 4;
      int n = i & 15;
      out[((size_t)(nt * 16 + n) * COUT + o) * HW + pos] = 0.0f;
    }
    return;
  }

  int ph = xo * STRIDE + (selh ? selh[pos] : 0);
  if (ph > HC - 1) ph = HC - 1;
  int pw = yo * STRIDE + (selw ? selw[pos] : 0);
  if (pw > WC - 1) pw = WC - 1;

  const float* __restrict__ inBlk = in + (size_t)(nt * 16) * CIN * HIN * WIN;

  v8f acc[TPW];
#pragma unroll
  for (int t = 0; t < TPW; ++t) {
    v8f z = {};
    acc[t] = z;
  }

  for (int ch = 0; ch < NCHUNK; ++ch) {
    const int kbase = ch * KC;
    if (ch) __syncthreads();  // protect LDS before overwrite

    // Cooperative im2col stage: lds[n*LROW + kk] = patch[n][kbase+kk].
    // All divisors are compile-time constants.
    for (int idx = tid; idx < 16 * KC; idx += 256) {
      const int n = idx / KC;
      const int kk = idx - n * KC;
      const int kp = kbase + kk;
      float v = 0.0f;
      if (kp < K) {
        const int c = kp / K2;
        const int rem = kp - c * K2;
        const int i = rem / KSZ;
        const int j = rem - i * KSZ;
        v = inBlk[((size_t)n * CIN + c) * (HIN * WIN) + (ph + i) * WIN +
                  (pw + j)];
      }
      lds[n * LROW + kk] = v;
    }
    __syncthreads();

#pragma unroll
    for (int t = 0; t < TPW; ++t) {
      const int mt = wave + t * NW;
      if (mt < MT) {  // wave-uniform -> branch-skipped, EXEC all-1s at WMMA
        const float* __restrict__ aRow =
            Wp + (size_t)(mt * 16 + l16) * KP + kbase;
        const float* __restrict__ bRow = lds + l16 * LROW;
        v8f c = acc[t];
#pragma unroll 8
        for (int kk = 0; kk < KC; kk += 4) {
          const int k0 = kk + 2 * half;
          v2f a = *(const v2f*)(aRow + k0);    // global_load_b64 (L2-resident)
          v2f bb = *(const v2f*)(bRow + k0);   // ds_load_b64
          c = __builtin_amdgcn_wmma_f32_16x16x4_f32(false, a, false, bb,
                                                    (short)0, c, false, false);
        }
        acc[t] = c;
      }
    }
  }

  // Epilogue: bias + mask + ReLU + store.
#pragma unroll
  for (int t = 0; t < TPW; ++t) {
    const int mt = wave + t * NW;
    if (mt < MT) {
#pragma unroll
      for (int r = 0; r < 8; ++r) {
        const int o = mt * 16 + r + 8 * half;
        if (COUT % 16 == 0 || o < COUT) {
          float v = (acc[t][r] + bias[o]) * mval;
          if (RELU) v = fmaxf(v, 0.0f);
          out[((size_t)b * COUT + o) * HW + pos] = v;
        }
      }
    }
  }
}

extern "C" void kernel_launch(void* const* d_in, const int* in_sizes, int n_in,
                              void* d_out, int out_size, void* d_ws,
                              size_t ws_size, hipStream_t stream) {
  (void)in_sizes; (void)n_in; (void)out_size; (void)ws_size;

  const float* x = (const float*)d_in[0];
  const float* W1 = (const float*)d_in[1];
  const float* b1 = (const float*)d_in[2];
  const float* W2 = (const float*)d_in[3];
  const float* b2 = (const float*)d_in[4];
  const float* W3 = (const float*)d_in[5];
  const float* b3 = (const float*)d_in[6];
  const float* fcw = (const float*)d_in[7];
  const float* fcb = (const float*)d_in[8];
  const int* selh1 = (const int*)d_in[9];
  const int* selw1 = (const int*)d_in[10];
  const int* selh2 = (const int*)d_in[11];
  const int* selw2 = (const int*)d_in[12];
  const int* selh3 = (const int*)d_in[13];
  const int* selw3 = (const int*)d_in[14];
  const float* m1 = (const float*)d_in[15];
  const float* m2 = (const float*)d_in[16];
  const float* m3 = (const float*)d_in[17];

  const int B = 1024;
  float* h1 = (float*)d_ws;                   // 1024*200*196
  float* h2 = h1 + (size_t)B * 200 * 196;     // 1024*400*36
  float* h3 = h2 + (size_t)B * 400 * 36;      // 1024*800
  float* W1p = h3 + (size_t)B * 800;          // 208*76
  float* FCp = W1p + (size_t)208 * 76;        // 16*800

  // Weight padding (L2/L3 already 16/4-aligned -> used in place).
  pad_weights<<<(208 * 76 + 255) / 256, 256, 0, stream>>>(W1, W1p, 200, 75,
                                                          208, 76);
  pad_weights<<<(16 * 800 + 255) / 256, 256, 0, stream>>>(fcw, FCp, 10, 800,
                                                          16, 800);

  // L1: (1024,3,32,32) -> (1024,200,14,14), k=5 s=2, K=75
  sconv_wmma<3, 32, 32, 200, 5, 2, 14, 14, 1>
      <<<dim3(196, B / 16), 256, smemFor(75), stream>>>(x, W1p, b1, selh1,
                                                        selw1, m1, h1);
  // L2: -> (1024,400,6,6), k=3 s=2, K=1800
  sconv_wmma<200, 14, 14, 400, 3, 2, 6, 6, 1>
      <<<dim3(36, B / 16), 256, smemFor(1800), stream>>>(h1, W2, b2, selh2,
                                                         selw2, m2, h2);
  // L3: -> (1024,800,1,1), k=3 s=4, K=3600
  sconv_wmma<400, 6, 6, 800, 3, 4, 1, 1, 1>
      <<<dim3(1, B / 16), 256, smemFor(3600), stream>>>(h2, W3, b3, selh3,
                                                        selw3, m3, h3);
  // FC as 1x1 "conv": Cout=10 (padded to 16), K=800, no mask/sel, no ReLU.
  sconv_wmma<800, 1, 1, 10, 1, 1, 1, 1, 0>
      <<<dim3(1, B / 16), 256, smemFor(800), stream>>>(
          h3, FCp, fcb, nullptr, nullptr, nullptr, (float*)d_out);
}